// MultiheadAttention_54331336295020
// MI455X (gfx1250) — compile-verified
//
#include <hip/hip_runtime.h>
#include <hip/hip_bf16.h>
#include <math.h>

typedef float v2f __attribute__((ext_vector_type(2)));
typedef float v8f __attribute__((ext_vector_type(8)));

// D = A(16x4) * B(4x16) + C, all fp32, matrix pipe.
#define WMMA_F32(a, b, c) \
  __builtin_amdgcn_wmma_f32_16x16x4_f32(false, (a), false, (b), (short)0, (c), false, false)

// Async DMA: global -> LDS, 16B per lane, tracked by ASYNCcnt (CDNA5).
// GVS addressing: saddr (SGPR64 base) + 32-bit per-lane byte offset.
__device__ __forceinline__ void async_g2l_b128(unsigned lds_byte_off,
                                               const float* __restrict__ sbase,
                                               unsigned gbyte_off) {
  asm volatile("global_load_async_to_lds_b128 %0, %1, %2"
               :: "v"(lds_byte_off), "v"(gbyte_off), "s"(sbase) : "memory");
}
__device__ __forceinline__ void wait_async_le4() {
  asm volatile("s_wait_asynccnt 0x4" ::: "memory");
}
__device__ __forceinline__ void wait_async_0() {
  asm volatile("s_wait_asynccnt 0x0" ::: "memory");
}

// ---------------------------------------------------------------------------
// GEMM: Y[M,N] = X[M,K] * W[N,K]^T + bias[N]
// rows_per_batch > 0 : scatter output into head-split layout [B, H=8, rows, 64]
// rows_per_batch == 0: flat row-major [M, N]
// 64x64 tile / workgroup, 8 waves (each 16x32 via two 16x16 accumulators).
// Double-buffered async global->LDS staging of 64x32 K-chunks.
// ---------------------------------------------------------------------------
__global__ __launch_bounds__(256) void proj_gemm_kernel(
    const float* __restrict__ X, const float* __restrict__ W,
    const float* __restrict__ bias, float* __restrict__ Y,
    int M, int K, int N, int rows_per_batch)
{
  constexpr int ST   = 36;        // 32 + 4 pad: frag bank = (36*m + k) % 64, conflict-free
  constexpr int BUFE = 64 * ST;   // elements per (tile, buffer)
  __shared__ float Xs[2 * BUFE];
  __shared__ float Ws[2 * BUFE];

  const int tid  = threadIdx.x;
  const int lane = tid & 31;
  const int wv   = tid >> 5;
  const int lo   = lane & 15;
  const int hi   = lane >> 4;
  const int m0   = blockIdx.x * 64;
  const int n0   = blockIdx.y * 64;
  const int mt   = (wv >> 1) * 16;  // 0,16,32,48
  const int nt   = (wv & 1) * 32;   // 0,32

  // LDS byte offsets (flat shared address truncates to LDS offset, ISA 10.2).
  const unsigned xb = (unsigned)(uintptr_t)(&Xs[0]);
  const unsigned wb = (unsigned)(uintptr_t)(&Ws[0]);

  // Issue one K-chunk's DMA into buffer `buf`: X and W 64x32 tiles.
  auto issue_chunk = [&](int kc, int buf) {
    // 64 rows x 32 cols = 512 float4 per tile; 2 per thread per tile.
    for (int idx = tid; idx < 512; idx += 256) {
      int r  = idx >> 3;         // 0..63
      int c4 = (idx & 7) * 4;    // 0..28
      unsigned lofX = xb + (unsigned)((buf * BUFE + r * ST + c4) * 4);
      unsigned lofW = wb + (unsigned)((buf * BUFE + r * ST + c4) * 4);
      unsigned gofX = (unsigned)(((m0 + r) * K + kc + c4) * 4);
      unsigned gofW = (unsigned)(((n0 + r) * K + kc + c4) * 4);
      async_g2l_b128(lofX, X, gofX);
      async_g2l_b128(lofW, W, gofW);
    }
  };

  v8f acc0 = {};
  v8f acc1 = {};

  const int nchunks = K >> 5;     // K/32
  issue_chunk(0, 0);
  int cur = 0;
  for (int c = 0; c < nchunks; ++c) {
    if (c + 1 < nchunks) {
      issue_chunk((c + 1) << 5, cur ^ 1);  // stream next chunk behind compute
      wait_async_le4();                    // only chunk c guaranteed complete
    } else {
      wait_async_0();
    }
    __syncthreads();

    const float* Xc = &Xs[cur * BUFE];
    const float* Wc = &Ws[cur * BUFE];
#pragma unroll
    for (int kk = 0; kk < 32; kk += 4) {
      // A frag: lane m holds X[m, kk+2*hi .. +1]
      v2f a  = *(const v2f*)&Xc[(mt + lo) * ST + kk + 2 * hi];
      // B frag: B[k,n] = W[n,k]; lane n holds W[n, kk+2*hi .. +1]
      v2f b0 = *(const v2f*)&Wc[(nt + lo) * ST + kk + 2 * hi];
      v2f b1 = *(const v2f*)&Wc[(nt + 16 + lo) * ST + kk + 2 * hi];
      acc0 = WMMA_F32(a, b0, acc0);
      acc1 = WMMA_F32(a, b1, acc1);
    }
    __syncthreads();  // all waves done with buf before it is re-filled
    cur ^= 1;
  }

  // Epilogue: bias + store.  C layout: lane l, vgpr r -> (m = r + 8*hi, n = l%16).
  float bn0 = bias ? bias[n0 + nt + lo]      : 0.0f;
  float bn1 = bias ? bias[n0 + nt + 16 + lo] : 0.0f;
#pragma unroll
  for (int r = 0; r < 8; ++r) {
    int m  = m0 + mt + r + 8 * hi;
    int c0 = n0 + nt + lo;
    int c1 = c0 + 16;
    float y0 = acc0[r] + bn0;
    float y1 = acc1[r] + bn1;
    if (rows_per_batch > 0) {
      int b  = m / rows_per_batch;
      int rr = m - b * rows_per_batch;
      Y[(((size_t)(b * 8 + (c0 >> 6))) * rows_per_batch + rr) * 64 + (c0 & 63)] = y0;
      Y[(((size_t)(b * 8 + (c1 >> 6))) * rows_per_batch + rr) * 64 + (c1 & 63)] = y1;
    } else {
      Y[(size_t)m * N + c0] = y0;
      Y[(size_t)m * N + c1] = y1;
    }
  }
}

// ---------------------------------------------------------------------------
// Attention core. One workgroup (4 waves) owns a (b, h, 16-query) tile.
//   Phase 1: S = Q K^T / 8 into LDS (16 x 4096 f32 = 256 KB, fits 320 KB WGP LDS)
//   Phase 2: exact softmax in LDS; stream normalized A to HBM exactly once
//   Phase 3: ctx = A @ V from LDS-resident A (V served from L2)
// ---------------------------------------------------------------------------
#define SPAD 4100  // 4096 + 4 pad: bank = 4*row + col -> conflict-free frag reads

__global__ __launch_bounds__(128) void attn_kernel(
    const float* __restrict__ Qws, const float* __restrict__ Kws,
    const float* __restrict__ Vws, float* __restrict__ Aout,
    float* __restrict__ Cws)
{
  extern __shared__ float lds[];
  float* Ssc = lds;              // 16 x SPAD scores
  float* Qs  = lds + 16 * SPAD;  // 16 x 68 Q tile

  const int tid  = threadIdx.x;
  const int lane = tid & 31;
  const int wv   = tid >> 5;     // 0..3
  const int lo   = lane & 15;
  const int hi   = lane >> 4;
  const int q0   = blockIdx.x * 16;
  const int h    = blockIdx.y;
  const int b    = blockIdx.z;
  const int bh   = b * 8 + h;

  const float* Qh = Qws + (size_t)bh * 512 * 64;
  const float* Kh = Kws + (size_t)bh * 4096 * 64;
  const float* Vh = Vws + (size_t)bh * 4096 * 64;

  // Load Q tile [16 x 64] into padded LDS.
  for (int idx = tid; idx < 16 * 64; idx += 128) {
    int m = idx >> 6, k = idx & 63;
    Qs[m * 68 + k] = Qh[(size_t)(q0 + m) * 64 + k];
  }
  __syncthreads();

  // ---- Phase 1: scores. Wave wv covers key columns [wv*1024, wv*1024+1024).
  for (int t = 0; t < 64; ++t) {
    int n0 = wv * 1024 + t * 16;
    v8f acc = {};
#pragma unroll
    for (int k = 0; k < 64; k += 4) {
      v2f a = *(const v2f*)&Qs[lo * 68 + k + 2 * hi];
      // B[k,n] = K[n,k]: lane n loads K[n0+n, k+2*hi .. +1]
      v2f bf = *(const v2f*)&Kh[(size_t)(n0 + lo) * 64 + k + 2 * hi];
      acc = WMMA_F32(a, bf, acc);
    }
#pragma unroll
    for (int r = 0; r < 8; ++r)
      Ssc[(r + 8 * hi) * SPAD + n0 + lo] = acc[r] * 0.125f;  // 1/sqrt(64)
  }
  __syncthreads();

  // ---- Phase 2: softmax per row; 4 rows per wave; write A once, normalized.
  for (int rw = 0; rw < 4; ++rw) {
    int row = wv * 4 + rw;
    float* Sr = &Ssc[row * SPAD];

    float mx = -3.4e38f;
    for (int c = lane; c < 4096; c += 32) mx = fmaxf(mx, Sr[c]);
    for (int o = 16; o > 0; o >>= 1) mx = fmaxf(mx, __shfl_xor(mx, o, 32));

    float sum = 0.0f;
    for (int c = lane; c < 4096; c += 32) {
      float p = __expf(Sr[c] - mx);
      Sr[c] = p;
      sum += p;
    }
    for (int o = 16; o > 0; o >>= 1) sum += __shfl_xor(sum, o, 32);
    float inv = 1.0f / sum;

    float* Ar = Aout + ((size_t)bh * 512 + q0 + row) * 4096;
    for (int j = 0; j < 32; ++j) {
      int c = (lane + 32 * j) * 4;
      float4 v = *(float4*)&Sr[c];
      v.x *= inv; v.y *= inv; v.z *= inv; v.w *= inv;
      *(float4*)&Sr[c] = v;   // keep normalized A in LDS for phase 3
      *(float4*)&Ar[c] = v;   // single HBM write of A
    }
  }
  __syncthreads();

  // ---- Phase 3: ctx = A @ V. Wave wv owns head features [wv*16, wv*16+16).
  {
    int nf = wv * 16 + lo;
    v8f acc = {};
    for (int k = 0; k < 4096; k += 4) {
      v2f a = *(const v2f*)&Ssc[lo * SPAD + k + 2 * hi];
      v2f bf;
      bf.x = Vh[(size_t)(k + 2 * hi)     * 64 + nf];
      bf.y = Vh[(size_t)(k + 2 * hi + 1) * 64 + nf];
      acc = WMMA_F32(a, bf, acc);
    }
#pragma unroll
    for (int r = 0; r < 8; ++r) {
      int m = r + 8 * hi;
      // ctx written flat [B, Nq, D] so the output projection reuses the GEMM.
      Cws[((size_t)(b * 512 + q0 + m)) * 512 + h * 64 + wv * 16 + lo] = acc[r];
    }
  }
}

// ---------------------------------------------------------------------------
extern "C" void kernel_launch(void* const* d_in, const int* in_sizes, int n_in,
                              void* d_out, int out_size, void* d_ws, size_t ws_size,
                              hipStream_t stream) {
  (void)in_sizes; (void)n_in; (void)out_size; (void)ws_size;

  const float* query = (const float*)d_in[0];
  const float* key   = (const float*)d_in[1];
  const float* value = (const float*)d_in[2];
  const float* Wq    = (const float*)d_in[3];
  const float* bq    = (const float*)d_in[4];
  const float* Wk    = (const float*)d_in[5];
  const float* bk    = (const float*)d_in[6];
  const float* Wv    = (const float*)d_in[7];
  const float* bv    = (const float*)d_in[8];
  const float* Wo    = (const float*)d_in[9];
  const float* bo    = (const float*)d_in[10];

  float* out  = (float*)d_out;                       // [4,512,512]
  float* Aout = out + (size_t)4 * 512 * 512;         // [4,8,512,4096]

  float* ws  = (float*)d_ws;
  float* Qws = ws;                                   // [4,8,512,64]   4 MB
  float* Kws = Qws + (size_t)4 * 8 * 512 * 64;       // [4,8,4096,64] 32 MB
  float* Vws = Kws + (size_t)4 * 8 * 4096 * 64;      // [4,8,4096,64] 32 MB
  float* Cws = Vws + (size_t)4 * 8 * 4096 * 64;      // [4,512,512]    4 MB

  const int smem = (16 * SPAD + 16 * 68) * (int)sizeof(float);  // ~260.5 KB
  (void)hipFuncSetAttribute((const void*)attn_kernel,
                            hipFuncAttributeMaxDynamicSharedMemorySize, smem);

  // Q/K/V projections (head-split outputs).
  proj_gemm_kernel<<<dim3(2048 / 64, 512 / 64), 256, 0, stream>>>(
      query, Wq, bq, Qws, 2048, 512, 512, 512);
  proj_gemm_kernel<<<dim3(16384 / 64, 512 / 64), 256, 0, stream>>>(
      key, Wk, bk, Kws, 16384, 512, 512, 4096);
  proj_gemm_kernel<<<dim3(16384 / 64, 512 / 64), 256, 0, stream>>>(
      value, Wv, bv, Vws, 16384, 512, 512, 4096);

  // Attention core: scores -> softmax -> A (HBM) + ctx, all from LDS.
  attn_kernel<<<dim3(32, 8, 4), 128, smem, stream>>>(Qws, Kws, Vws, Aout, Cws);

  // Output projection (flat layout, straight into d_out).
  proj_gemm_kernel<<<dim3(2048 / 64, 512 / 64), 256, 0, stream>>>(
      Cws, Wo, bo, out, 2048, 512, 512, 0);
}